// GraphConvV_6451040878948
// MI455X (gfx1250) — compile-verified
//
#include <hip/hip_runtime.h>

// Problem constants (from reference): B=8, A=128, C=3, F1=F2=64, FILT=128
#define NB   8
#define NA   128
#define NC   3
#define NF   64          // per-tensor feature dim
#define FILT 128
#define KDIM (2*NF)      // 128 = concat feature dim
#define NROWS (NB*NA*NC) // 3072 GEMM rows

typedef __attribute__((ext_vector_type(4))) float v4f;
typedef __attribute__((ext_vector_type(2))) float v2f;
typedef __attribute__((ext_vector_type(8))) float v8f;

// ---------------------------------------------------------------------------
// Stage 1: s[bi*3 + c][f'] = sum_j adj[bi][j] * concat(v1,v2)[bi][j][c][f']
//          deg[bi]         = sum_j adj[bi][j]
// One block per (b,i): 96 threads, each owns one float4 (4 features) of the
// 384-wide per-(b,i) feature vector (3 c x 128). Streams 196 KB per block, 201 MB total.
// ---------------------------------------------------------------------------
__global__ void __launch_bounds__(96)
graphconv_reduce_j(const float* __restrict__ v1,
                   const float* __restrict__ v2,
                   const float* __restrict__ adj,
                   float* __restrict__ s,
                   float* __restrict__ deg)
{
    const int bi = blockIdx.x;            // 0 .. B*A-1
    const int t  = threadIdx.x;           // 0 .. 95

    __shared__ float lds_adj[NA];
    const float* adjrow = adj + (size_t)bi * NA;
    for (int j = t; j < NA; j += 96) lds_adj[j] = adjrow[j];
    __syncthreads();

    // thread -> tensor + feature slice
    const bool second = (t >= 48);
    const int  tt     = second ? (t - 48) : t;
    const int  p      = tt * 4;                         // feature offset in [0,192)
    const float* src  = (second ? v2 : v1)
                      + (size_t)bi * (size_t)(NA * NC * NF)   // (b,i) base: 128*192
                      + p;

    v4f acc = {};
    #pragma unroll 4
    for (int j = 0; j < NA; ++j) {
        const float aj = lds_adj[j];
        const v4f x = __builtin_nontemporal_load(
            reinterpret_cast<const v4f*>(src + (size_t)j * (NC * NF)));
        acc += aj * x;
    }

    // scatter into s: row = bi*3 + c, col = f (+64 for second tensor)
    const int c   = p / NF;
    const int f   = p % NF;
    const int col = f + (second ? NF : 0);
    *reinterpret_cast<v4f*>(s + ((size_t)bi * NC + c) * KDIM + col) = acc;

    if (t == 0) {
        float d = 0.0f;
        #pragma unroll
        for (int j = 0; j < NA; ++j) d += lds_adj[j];
        deg[bi] = d;
    }
}

// ---------------------------------------------------------------------------
// Stage 2: out[row][o] = relu( s[row][:] @ w[:][o] + deg[row/3]*bias[o] )
// M=3072, N=128, K=128.  One wave per 16x16 tile via V_WMMA_F32_16X16X4_F32,
// K handled as 32 chained K=4 WMMAs. Block = 256 threads = 8 waves = the 8
// N-tiles of one M-tile; grid = 192 M-tiles. EXEC is all-ones everywhere.
//
// Fragment layouts per CDNA5 ISA (CDNA5 ISA):
//   A 16x4 f32 : lane L, vgpr v -> A[M = L%16][K = v + 2*(L>=16)]
//   B 4x16 f32 : lane L, vgpr v -> B[K = v + 2*(L>=16)][N = L%16]
//   C/D 16x16  : lane L, vgpr r -> C[M = r + 8*(L>=16)][N = L%16]
// ---------------------------------------------------------------------------
__global__ void __launch_bounds__(256)
graphconv_gemm_wmma(const float* __restrict__ s,
                    const float* __restrict__ w,
                    const float* __restrict__ bias,
                    const float* __restrict__ deg,
                    float* __restrict__ out)
{
    const int lane  = threadIdx.x & 31;
    const int wave  = threadIdx.x >> 5;     // 0..7 -> N tile
    const int mtile = blockIdx.x;           // 0..191
    const int l16   = lane & 15;
    const int lhalf = lane >> 4;            // 0/1

    const int row0 = mtile * 16;
    const int col  = wave * 16 + l16;

    const float* arow = s + ((size_t)(row0 + l16)) * KDIM + 2 * lhalf;
    const float* bcol = w + (size_t)(2 * lhalf) * FILT + col;

    v8f acc = {};
    #pragma unroll
    for (int k = 0; k < KDIM; k += 4) {
        // A fragment: contiguous float2 of s row
        const v2f a = *reinterpret_cast<const v2f*>(arow + k);
        // B fragment: two rows of w, stride FILT
        v2f b;
        b.x = bcol[(size_t)k * FILT];
        b.y = bcol[(size_t)(k + 1) * FILT];
        acc = __builtin_amdgcn_wmma_f32_16x16x4_f32(
            /*neg_a=*/false, a, /*neg_b=*/false, b,
            /*c_mod=*/(short)0, acc, /*reuse_a=*/false, /*reuse_b=*/false);
    }

    const float bo = bias[col];
    #pragma unroll
    for (int r = 0; r < 8; ++r) {
        const int row = row0 + r + 8 * lhalf;
        float v = acc[r] + deg[(unsigned)row / 3u] * bo;
        v = fmaxf(v, 0.0f);
        out[(size_t)row * FILT + col] = v;
    }
}

// ---------------------------------------------------------------------------
extern "C" void kernel_launch(void* const* d_in, const int* in_sizes, int n_in,
                              void* d_out, int out_size, void* d_ws, size_t ws_size,
                              hipStream_t stream) {
    const float* v1   = (const float*)d_in[0];   // [8,128,128,3,64]
    const float* v2   = (const float*)d_in[1];   // [8,128,128,3,64]
    const float* adj  = (const float*)d_in[2];   // [8,128,128]
    const float* w    = (const float*)d_in[3];   // [128,128]
    const float* bias = (const float*)d_in[4];   // [128]
    float*       out  = (float*)d_out;           // [8,128,3,128]

    float* s   = (float*)d_ws;                   // [3072,128] f32  (1.5 MB)
    float* deg = s + (size_t)NROWS * KDIM;       // [1024] f32

    graphconv_reduce_j<<<NB * NA, 96, 0, stream>>>(v1, v2, adj, s, deg);
    graphconv_gemm_wmma<<<NROWS / 16, 256, 0, stream>>>(s, w, bias, deg, out);
}